// MotionCorruptionSimulator_42760694399220
// MI455X (gfx1250) — compile-verified
//
#include <hip/hip_runtime.h>
#include <math.h>

// Problem constants (match reference)
#define BB 32
#define CC 8
#define HH 320
#define WW 320
#define NPIX (HH * WW)            // 102400, divisible by 256 -> no bounds checks
#define PLANE (HH * WW)           // per-channel plane
#define IMGSZ (CC * HH * WW)      // per-batch image
#define DEG2RAD 0.017453292519943295f

// Small-angle sin/cos (|a| <= ~0.35 rad is ample; input is <= 0.0873 rad).
// Horner in t = a^2; truncation error << f32 ulp for this range.
// No range reduction -> no libm Payne-Hanek SALU blob in the kernel.
__device__ __forceinline__ float small_sinf(float a, float t) {
    // a * (1 - t/6 + t^2/120 - t^3/5040)
    float p = fmaf(t, -1.9841270e-04f, 8.3333333e-03f);   // t/5040, 1/120
    p = fmaf(t, p, -1.6666667e-01f);                      // -1/6
    p = fmaf(t, p, 1.0f);
    return a * p;
}
__device__ __forceinline__ float small_cosf(float t) {
    // 1 - t/2 + t^2/24 - t^3/720
    float p = fmaf(t, -1.3888889e-03f, 4.1666667e-02f);   // 1/720, 1/24
    p = fmaf(t, p, -0.5f);
    p = fmaf(t, p, 1.0f);
    return p;
}

// Main warp kernel: one thread per (b, y, x) pixel, unrolled over 8 channels.
// blockIdx.y = b  -> per-batch params are block-uniform (scalarizable loads + math).
__global__ __launch_bounds__(256) void motion_warp_kernel(
    const float* __restrict__ img,
    const float* __restrict__ ang_deg,
    const float* __restrict__ txv,
    const float* __restrict__ tyv,
    float* __restrict__ out)
{
    const int b   = blockIdx.y;
    const int pix = blockIdx.x * 256 + threadIdx.x;   // 0 .. NPIX-1 exactly
    const int y   = pix / WW;
    const int x   = pix - y * WW;

    // Per-batch transform (uniform within block; polynomial sin/cos)
    const float a   = ang_deg[b] * DEG2RAD;
    const float t   = a * a;
    const float s   = small_sinf(a, t);
    const float c   = small_cosf(t);
    const float txn = txv[b] * (2.0f / (float)WW);
    const float tyn = tyv[b] * (2.0f / (float)HH);

    // Normalized grid coords: linspace(-1, 1, N)
    const float gxb = -1.0f + (float)x * (2.0f / (float)(WW - 1));
    const float gyb = -1.0f + (float)y * (2.0f / (float)(HH - 1));
    const float gx  = c * gxb - s * gyb + txn;
    const float gy  = s * gxb + c * gyb + tyn;

    // Map to pixel space with border clamp
    float ix = (gx + 1.0f) * (0.5f * (float)(WW - 1));
    float iy = (gy + 1.0f) * (0.5f * (float)(HH - 1));
    ix = fminf(fmaxf(ix, 0.0f), (float)(WW - 1));
    iy = fminf(fmaxf(iy, 0.0f), (float)(HH - 1));

    const float x0f = floorf(ix);
    const float y0f = floorf(iy);
    const int   x0  = (int)x0f;
    const int   y0  = (int)y0f;
    const int   x1  = min(x0 + 1, WW - 1);
    const int   y1  = min(y0 + 1, HH - 1);
    const float wx  = ix - x0f;
    const float wy  = iy - y0f;

    const float w00 = (1.0f - wx) * (1.0f - wy);
    const float w01 = wx * (1.0f - wy);
    const float w10 = (1.0f - wx) * wy;
    const float w11 = wx * wy;

    const int i00 = y0 * WW + x0;
    const int i01 = y0 * WW + x1;
    const int i10 = y1 * WW + x0;
    const int i11 = y1 * WW + x1;

    const float* __restrict__ ip = img + b * IMGSZ;   // < 2^31, int addressing
    float*       __restrict__ op = out + b * IMGSZ;

    // Same sample coords for all channels: amortize geometry over C=8.
    // Loads: default RT hint (whole 105MB input lives in the 192MB L2).
    // Stores: non-temporal (write-once stream, don't evict the input).
    #pragma unroll
    for (int ch = 0; ch < CC; ++ch) {
        const float* p = ip + ch * PLANE;
        const float v = fmaf(p[i00], w00,
                        fmaf(p[i01], w01,
                        fmaf(p[i10], w10, p[i11] * w11)));
        __builtin_nontemporal_store(v, op + ch * PLANE + pix);
    }
}

// Tail outputs: a = deg->rad, translations = [tx0,ty0, tx1,ty1, ...]
__global__ __launch_bounds__(32) void motion_tail_kernel(
    const float* __restrict__ ang_deg,
    const float* __restrict__ txv,
    const float* __restrict__ tyv,
    float* __restrict__ tail)   // points at out + B*C*H*W
{
    const int i = threadIdx.x;
    if (i < BB) {
        tail[i] = ang_deg[i] * DEG2RAD;
        tail[BB + 2 * i + 0] = txv[i];
        tail[BB + 2 * i + 1] = tyv[i];
    }
}

extern "C" void kernel_launch(void* const* d_in, const int* in_sizes, int n_in,
                              void* d_out, int out_size, void* d_ws, size_t ws_size,
                              hipStream_t stream) {
    (void)in_sizes; (void)n_in; (void)out_size; (void)d_ws; (void)ws_size;

    const float* img = (const float*)d_in[0];   // clean_image [32,8,320,320] f32
    const float* ang = (const float*)d_in[1];   // angles_deg  [32] f32
    const float* tx  = (const float*)d_in[2];   // tx          [32] f32
    const float* ty  = (const float*)d_in[3];   // ty          [32] f32
    float* out = (float*)d_out;

    dim3 grid(NPIX / 256, BB);   // 400 x 32 blocks, 256 threads (8 wave32s)
    motion_warp_kernel<<<grid, 256, 0, stream>>>(img, ang, tx, ty, out);
    motion_tail_kernel<<<1, 32, 0, stream>>>(ang, tx, ty,
                                             out + (size_t)BB * IMGSZ);
}